// BiLSTM_CRF_30846455120522
// MI455X (gfx1250) — compile-verified
//
#include <hip/hip_runtime.h>
#include <hip/hip_bf16.h>
#include <math.h>
#include <stdint.h>

// Problem constants (match reference)
#define S_LEN 4096
#define E_DIM 512
#define H_DIM 512   // per-direction hidden
#define FH    2048  // 4*H
#define HIDC  1024  // 2*H (concat)
#define T_TAGS 32

typedef float v2f __attribute__((ext_vector_type(2)));
typedef float v8f __attribute__((ext_vector_type(8)));

__device__ __forceinline__ float sigmoidf_(float x) { return 1.0f / (1.0f + __expf(-x)); }

// Low 32 bits of a generic pointer to LDS == wave-relative LDS byte address.
__device__ __forceinline__ uint32_t lds_addr32(const void* p) {
    return (uint32_t)(uintptr_t)p;
}

// ---------------------------------------------------------------------------
// Kernel 1: fused embedding gather + input projection, fp32 WMMA.
//   out[m][n] = sum_k emb[ids[m']][k] * wih[n][k] + bias[n]
//   m' = reverse ? S-1-m : m   (backward direction consumes reversed sequence)
// One wave per 16x16 output tile; K-loop in steps of 4 (V_WMMA_F32_16X16X4_F32).
// A-frag: lane l (0..15) = row m0+l, K={0,1}; lanes 16..31 same row, K={2,3}.
// B-frag: lane l = col n0+l (i.e. W row n0+l), same K split.
// C/D:    VGPR r = row m0 + r + 8*(lane>=16), col n0 + (lane&15).
// ---------------------------------------------------------------------------
__global__ void xproj_kernel(const int* __restrict__ ids,
                             const float* __restrict__ emb,
                             const float* __restrict__ wih,
                             const float* __restrict__ bias,
                             float* __restrict__ out,
                             int reverse) {
    const int lane = threadIdx.x;       // 0..31 (wave32)
    const int l    = lane & 15;
    const int half = lane >> 4;
    const int n0 = blockIdx.x * 16;     // gate-row tile
    const int m0 = blockIdx.y * 16;     // sequence tile

    const int m = m0 + l;
    const int s = reverse ? (S_LEN - 1 - m) : m;
    const float* __restrict__ arow = emb + (size_t)ids[s] * E_DIM;
    const float* __restrict__ brow = wih + (size_t)(n0 + l) * E_DIM;

    v8f c = {};
#pragma unroll 8
    for (int k = 0; k < E_DIM; k += 4) {
        v2f a = *(const v2f*)(arow + k + 2 * half);
        v2f b = *(const v2f*)(brow + k + 2 * half);
        c = __builtin_amdgcn_wmma_f32_16x16x4_f32(false, a, false, b,
                                                  (short)0, c, false, false);
    }
    const float bn = bias[n0 + l];
#pragma unroll
    for (int r = 0; r < 8; ++r) {
        const int row = m0 + r + 8 * half;
        out[(size_t)row * FH + n0 + l] = c[r] + bn;
    }
}

// ---------------------------------------------------------------------------
// Kernel 2: the serial LSTM scan, one 1024-thread workgroup per direction.
// Per step: gates[j] = xp[t][j] + whh[j][:] . h   (j in [0,2048))
//   thread tid < 512  : rows tid (i-gate) and 1024+tid (g-gate)
//   thread tid >= 512 : rows tid (f-gate) and 1024+tid (o-gate)
// h lives in LDS; c is a private register of threads 0..511.
// xp[t] gate rows are double-buffered into LDS with GLOBAL_LOAD_ASYNC_TO_LDS
// (ASYNCcnt-tracked), hiding the per-step input fetch behind the matvec.
// Backward direction writes h at position S-1-t (pre-un-reversed).
// ---------------------------------------------------------------------------
__global__ void lstm_kernel(const float* __restrict__ xp_f,
                            const float* __restrict__ xp_b,
                            const float* __restrict__ whh_f,
                            const float* __restrict__ whh_b,
                            float* __restrict__ hf_out,
                            float* __restrict__ hb_out) {
    const int dir = blockIdx.x;
    const float* __restrict__ xp   = dir ? xp_b  : xp_f;
    const float* __restrict__ whh  = dir ? whh_b : whh_f;
    float* __restrict__ hout       = dir ? hb_out : hf_out;

    __shared__ __align__(16) float h_s[H_DIM];
    __shared__ float g0_s[1024];
    __shared__ float g1_s[1024];
    __shared__ __align__(16) float xbuf[2][FH];   // async double buffer (16 KB)

    const int tid = threadIdx.x;        // 0..1023
    const int j0 = tid;
    const int j1 = 1024 + tid;
    const float4* __restrict__ w0 = reinterpret_cast<const float4*>(whh + (size_t)j0 * H_DIM);
    const float4* __restrict__ w1 = reinterpret_cast<const float4*>(whh + (size_t)j1 * H_DIM);
    // Warm L2 with this thread's weight rows (re-read 4096x).
    __builtin_prefetch(w0, 0, 3);
    __builtin_prefetch(w1, 0, 3);

    // Each thread async-copies bytes [8*tid, 8*tid+8) of the step's xp row.
    const uint32_t lds0 = lds_addr32(&xbuf[0][2 * tid]);
    const uint32_t lds1 = lds_addr32(&xbuf[1][2 * tid]);
    const unsigned long long gbase =
        (unsigned long long)(uintptr_t)xp + 8ull * (unsigned)tid;

    float cst = 0.0f;                   // cell state, owned by tid < 512
    if (tid < H_DIM) h_s[tid] = 0.0f;

    // Kick off t = 0 row into buffer 0.
    asm volatile("global_load_async_to_lds_b64 %0, %1, off"
                 :: "v"(lds0), "v"(gbase) : "memory");

    for (int t = 0; t < S_LEN; ++t) {
        if (t + 1 < S_LEN) {
            // Prefetch next step's row into the other buffer, then wait for
            // the *older* outstanding async op (this step's buffer).
            const uint32_t ldsn = ((t + 1) & 1) ? lds1 : lds0;
            const unsigned long long gan =
                gbase + (unsigned long long)(t + 1) * (4ull * FH);
            asm volatile("global_load_async_to_lds_b64 %0, %1, off"
                         :: "v"(ldsn), "v"(gan) : "memory");
            asm volatile("s_wait_asynccnt 0x1" ::: "memory");
        } else {
            asm volatile("s_wait_asynccnt 0x0" ::: "memory");
        }
        __syncthreads();                // all waves' async data + h_s visible

        const float* __restrict__ xb = xbuf[t & 1];
        float acc0 = xb[j0];
        float acc1 = xb[j1];
        const float4* __restrict__ h4 = reinterpret_cast<const float4*>(h_s);
#pragma unroll 4
        for (int kk = 0; kk < H_DIM / 4; ++kk) {
            const float4 hv = h4[kk];
            const float4 a  = w0[kk];
            const float4 b  = w1[kk];
            acc0 = fmaf(a.x, hv.x, fmaf(a.y, hv.y, fmaf(a.z, hv.z, fmaf(a.w, hv.w, acc0))));
            acc1 = fmaf(b.x, hv.x, fmaf(b.y, hv.y, fmaf(b.z, hv.z, fmaf(b.w, hv.w, acc1))));
        }
        g0_s[tid] = acc0;
        g1_s[tid] = acc1;
        __syncthreads();
        if (tid < H_DIM) {
            const float ig = sigmoidf_(g0_s[tid]);
            const float fg = sigmoidf_(g0_s[tid + H_DIM]);
            const float gg = tanhf(g1_s[tid]);
            const float og = sigmoidf_(g1_s[tid + H_DIM]);
            cst = fg * cst + ig * gg;
            const float h = og * tanhf(cst);
            h_s[tid] = h;
            const int pos = dir ? (S_LEN - 1 - t) : t;
            hout[(size_t)pos * H_DIM + tid] = h;
        }
        __syncthreads();
    }
}

// ---------------------------------------------------------------------------
// Kernel 3: emissions GEMM, fp32 WMMA.
//   em[s][t] = concat(h_f,h_b)[s] . lin_w[t] + lin_b[t]
// K-loop split into the h_f half and the h_b half (straight strided streams).
// ---------------------------------------------------------------------------
__global__ void emis_kernel(const float* __restrict__ h_f,
                            const float* __restrict__ h_b,
                            const float* __restrict__ lin_w,
                            const float* __restrict__ lin_b,
                            float* __restrict__ em) {
    const int lane = threadIdx.x;
    const int l    = lane & 15;
    const int half = lane >> 4;
    const int n0 = blockIdx.x * 16;     // tag tile (0 or 16)
    const int m0 = blockIdx.y * 16;

    const int m = m0 + l;
    const float* __restrict__ brow = lin_w + (size_t)(n0 + l) * HIDC;
    const float* __restrict__ af   = h_f + (size_t)m * H_DIM;
    const float* __restrict__ ab   = h_b + (size_t)m * H_DIM;

    v8f c = {};
#pragma unroll 8
    for (int k = 0; k < H_DIM; k += 4) {
        v2f a = *(const v2f*)(af + k + 2 * half);
        v2f b = *(const v2f*)(brow + k + 2 * half);
        c = __builtin_amdgcn_wmma_f32_16x16x4_f32(false, a, false, b,
                                                  (short)0, c, false, false);
    }
#pragma unroll 8
    for (int k = 0; k < H_DIM; k += 4) {
        v2f a = *(const v2f*)(ab + k + 2 * half);
        v2f b = *(const v2f*)(brow + H_DIM + k + 2 * half);
        c = __builtin_amdgcn_wmma_f32_16x16x4_f32(false, a, false, b,
                                                  (short)0, c, false, false);
    }
    const float bn = lin_b[n0 + l];
#pragma unroll
    for (int r = 0; r < 8; ++r) {
        const int row = m0 + r + 8 * half;
        em[(size_t)row * T_TAGS + n0 + l] = c[r] + bn;
    }
}

// ---------------------------------------------------------------------------
// Kernel 4: CRF gold score + forward algorithm (logZ). One wave32: lane == tag.
// ---------------------------------------------------------------------------
__global__ void crf_kernel(const float* __restrict__ em,
                           const int* __restrict__ target,
                           const float* __restrict__ start_trans,
                           const float* __restrict__ end_trans,
                           const float* __restrict__ trans,
                           float* __restrict__ out) {
    __shared__ float tr_s[T_TAGS * T_TAGS];
    __shared__ float alpha_s[T_TAGS];
    __shared__ float red_s[T_TAGS];
    __shared__ float score_s;

    const int lane = threadIdx.x;       // 0..31
    for (int i = lane; i < T_TAGS * T_TAGS; i += 32) tr_s[i] = trans[i];
    __syncthreads();

    // ---- gold path score (strided over lanes, reduced in LDS) ----
    float part = 0.0f;
    for (int s = lane; s < S_LEN; s += 32) part += em[(size_t)s * T_TAGS + target[s]];
    for (int s = lane + 1; s < S_LEN; s += 32)
        part += tr_s[target[s - 1] * T_TAGS + target[s]];
    red_s[lane] = part;
    __syncthreads();
    if (lane == 0) {
        float sc = start_trans[target[0]] + end_trans[target[S_LEN - 1]];
        for (int i = 0; i < 32; ++i) sc += red_s[i];
        score_s = sc;
    }

    // ---- forward algorithm: alpha[j] per lane ----
    float alpha = start_trans[lane] + em[lane];   // em[0][lane]
    for (int t = 1; t < S_LEN; ++t) {
        alpha_s[lane] = alpha;
        __syncthreads();
        float mx = -INFINITY;
#pragma unroll
        for (int i = 0; i < T_TAGS; ++i)
            mx = fmaxf(mx, alpha_s[i] + tr_s[i * T_TAGS + lane]);
        float sum = 0.0f;
#pragma unroll
        for (int i = 0; i < T_TAGS; ++i)
            sum += __expf(alpha_s[i] + tr_s[i * T_TAGS + lane] - mx);
        alpha = mx + __logf(sum) + em[(size_t)t * T_TAGS + lane];
        __syncthreads();
    }
    alpha += end_trans[lane];
    alpha_s[lane] = alpha;
    __syncthreads();
    if (lane == 0) {
        float mx = -INFINITY;
        for (int i = 0; i < 32; ++i) mx = fmaxf(mx, alpha_s[i]);
        float sum = 0.0f;
        for (int i = 0; i < 32; ++i) sum += __expf(alpha_s[i] - mx);
        out[0] = (mx + __logf(sum)) - score_s;   // logZ - score
    }
}

// ---------------------------------------------------------------------------
extern "C" void kernel_launch(void* const* d_in, const int* in_sizes, int n_in,
                              void* d_out, int out_size, void* d_ws, size_t ws_size,
                              hipStream_t stream) {
    (void)in_sizes; (void)n_in; (void)out_size; (void)ws_size;

    const int*   ids         = (const int*)  d_in[0];
    const int*   target      = (const int*)  d_in[1];
    const float* emb         = (const float*)d_in[2];
    const float* wih_f       = (const float*)d_in[3];
    const float* whh_f       = (const float*)d_in[4];
    const float* b_f         = (const float*)d_in[5];
    const float* wih_b       = (const float*)d_in[6];
    const float* whh_b       = (const float*)d_in[7];
    const float* b_b         = (const float*)d_in[8];
    const float* lin_w       = (const float*)d_in[9];
    const float* lin_b       = (const float*)d_in[10];
    const float* start_trans = (const float*)d_in[11];
    const float* end_trans   = (const float*)d_in[12];
    const float* trans       = (const float*)d_in[13];

    // Workspace layout (fp32), ~85 MB total.
    float* ws   = (float*)d_ws;
    size_t o    = 0;
    float* xp_f = ws + o; o += (size_t)S_LEN * FH;     // 8,388,608
    float* xp_b = ws + o; o += (size_t)S_LEN * FH;
    float* h_f  = ws + o; o += (size_t)S_LEN * H_DIM;  // 2,097,152
    float* h_b  = ws + o; o += (size_t)S_LEN * H_DIM;
    float* em   = ws + o; o += (size_t)S_LEN * T_TAGS; // 131,072

    // 1) input projections (embedding gather fused), both directions
    dim3 gproj(FH / 16, S_LEN / 16);               // (128, 256) tiles, 1 wave each
    xproj_kernel<<<gproj, 32, 0, stream>>>(ids, emb, wih_f, b_f, xp_f, 0);
    xproj_kernel<<<gproj, 32, 0, stream>>>(ids, emb, wih_b, b_b, xp_b, 1);

    // 2) serial bidirectional LSTM scan (2 concurrent workgroups)
    lstm_kernel<<<2, 1024, 0, stream>>>(xp_f, xp_b, whh_f, whh_b, h_f, h_b);

    // 3) emissions
    dim3 gem(T_TAGS / 16, S_LEN / 16);             // (2, 256)
    emis_kernel<<<gem, 32, 0, stream>>>(h_f, h_b, lin_w, lin_b, em);

    // 4) CRF NLL -> scalar
    crf_kernel<<<1, 32, 0, stream>>>(em, target, start_trans, end_trans, trans,
                                     (float*)d_out);
}